// DevignBaseline_42494406427516
// MI455X (gfx1250) — compile-verified
//
#include <hip/hip_runtime.h>

typedef __attribute__((ext_vector_type(2))) float v2f;
typedef __attribute__((ext_vector_type(8))) float v8f;

#define HID 128
#define G3  384
#define BSTRIDE 132   // LDS col stride (floats): bank = (4*m + k0) % 64 -> conflict-free

// ---------------------------------------------------------------------------
// Blocked fp32 WMMA GEMM:  out[N,J] = in[N,128] @ W (+ bias)
//   TRANSPOSE_W: W is [J,128], out = in @ W.T   (torch Linear convention)
//   else:        W is [128,J], out = in @ W
//   GATHER:      A row base = in + gidx[row]*128  (fused embedding lookup)
//
// Block = 8 waves = 128 rows x 64 cols macro-tile.
// B tile (64 cols x 128 k) staged in LDS once per block, transposed so the
// per-fragment read is a contiguous 8B ds load: Bt[c*BSTRIDE + k] = B[k][c].
// Each wave: 16 rows x 4 col-tiles; per k-step: 1 global b64 (A) +
// 4 ds b64 (B) + 4x V_WMMA_F32_16X16X4_F32.
// A-frag (16x4 f32): lane m=lane&15, half=lane>>4: a[v]=Arow[m][4s+2*half+v]
// C/D (16x16 f32): vgpr r -> row (r + 8*half), col = lane&15
// ---------------------------------------------------------------------------
template <bool TRANSPOSE_W, bool GATHER>
__global__ __launch_bounds__(256) void wmma_gemm_kernel(
    const float* __restrict__ in, const int* __restrict__ gidx,
    const float* __restrict__ W, const float* __restrict__ bias,
    float* __restrict__ out, int N, int J) {
  __shared__ float Bt[64 * BSTRIDE];

  const int tid  = threadIdx.x;
  const int lane = tid & 31;
  const int wave = tid >> 5;
  const int rowBase = blockIdx.x * 128;
  const int colBase = blockIdx.y * 64;

  // ---- stage B tile (64 cols x 128 k) into LDS, transposed ----
  {
    const int c = tid >> 2;            // 0..63  local col
    const int k0 = (tid & 3) * 32;     // 0,32,64,96
    if (TRANSPOSE_W) {
      // B[k][c] = W[colBase+c][k]: contiguous in k -> float4 loads
      const float* wr = W + (size_t)(colBase + c) * HID + k0;
      float* dstp = Bt + c * BSTRIDE + k0;
#pragma unroll
      for (int q = 0; q < 8; ++q) {
        float4 v = ((const float4*)wr)[q];
        ((float4*)dstp)[q] = v;        // 16B-aligned: BSTRIDE*4 = 528 = 33*16
      }
    } else {
      // B[k][c] = W[k][colBase+c]: strided gather
      float* dstp = Bt + c * BSTRIDE + k0;
#pragma unroll
      for (int q = 0; q < 32; ++q)
        dstp[q] = W[(size_t)(k0 + q) * J + (colBase + c)];
    }
  }
  __syncthreads();

  const int m15  = lane & 15;
  const int half = lane >> 4;
  const int half2 = half * 2;

  int rowA = rowBase + wave * 16 + m15;
  if (rowA >= N) rowA = N - 1;
  const float* arow;
  if (GATHER) arow = in + (size_t)gidx[rowA] * HID;
  else        arow = in + (size_t)rowA * HID;

  v8f acc0 = {}, acc1 = {}, acc2 = {}, acc3 = {};
#pragma unroll
  for (int s = 0; s < 32; ++s) {
    const int k = 4 * s + half2;
    v2f a;
    a.x = arow[k];
    a.y = arow[k + 1];
    const float* bp = Bt + m15 * BSTRIDE + k;   // tile 0 fragment base
    v2f b0; b0.x = bp[0 * 16 * BSTRIDE]; b0.y = bp[0 * 16 * BSTRIDE + 1];
    v2f b1; b1.x = bp[1 * 16 * BSTRIDE]; b1.y = bp[1 * 16 * BSTRIDE + 1];
    v2f b2; b2.x = bp[2 * 16 * BSTRIDE]; b2.y = bp[2 * 16 * BSTRIDE + 1];
    v2f b3; b3.x = bp[3 * 16 * BSTRIDE]; b3.y = bp[3 * 16 * BSTRIDE + 1];
    acc0 = __builtin_amdgcn_wmma_f32_16x16x4_f32(false, a, false, b0, (short)0, acc0, false, false);
    acc1 = __builtin_amdgcn_wmma_f32_16x16x4_f32(false, a, false, b1, (short)0, acc1, false, false);
    acc2 = __builtin_amdgcn_wmma_f32_16x16x4_f32(false, a, false, b2, (short)0, acc2, false, false);
    acc3 = __builtin_amdgcn_wmma_f32_16x16x4_f32(false, a, false, b3, (short)0, acc3, false, false);
  }

  // ---- epilogue: bias + store 4 tiles ----
  v8f accs[4] = {acc0, acc1, acc2, acc3};
#pragma unroll
  for (int t = 0; t < 4; ++t) {
    const int col = colBase + t * 16 + m15;
    const float bv = bias ? bias[col] : 0.0f;
#pragma unroll
    for (int r = 0; r < 8; ++r) {
      const int row = rowBase + wave * 16 + r + 8 * half;
      if (row < N) out[(size_t)row * J + col] = accs[t][r] + bv;
    }
  }
}

// ---------------------------------------------------------------------------
__global__ void fill_kernel(float* __restrict__ p, float val, long n) {
  long i = (long)blockIdx.x * blockDim.x + threadIdx.x;
  const long stride = (long)gridDim.x * blockDim.x;
  for (; i < n; i += stride) p[i] = val;
}

// One wave per edge; lane handles 4 consecutive feature columns (32*4 = 128).
// fp32 HW atomics -> global_atomic_add_f32 (L2 atomic units), bandwidth-bound.
__global__ void scatter_add_kernel(const float* __restrict__ m,
                                   const int* __restrict__ src,
                                   const int* __restrict__ dst,
                                   float* __restrict__ agg, int E) {
  const int lane = threadIdx.x & 31;
  int w = (int)((blockIdx.x * blockDim.x + threadIdx.x) >> 5);
  const int nw = (int)((gridDim.x * (long)blockDim.x) >> 5);
  for (; w < E; w += nw) {
    const int s = src[w];
    const int d = dst[w];
    const float4 v = ((const float4*)(m + (size_t)s * HID))[lane];
    float* ap = agg + (size_t)d * HID + lane * 4;
    atomicAdd(ap + 0, v.x);
    atomicAdd(ap + 1, v.y);
    atomicAdd(ap + 2, v.z);
    atomicAdd(ap + 3, v.w);
  }
}

// GRU cell elementwise (gate order r,z,n), in-place h update.
__global__ void gru_gate_kernel(const float* __restrict__ gi,
                                const float* __restrict__ gh,
                                float* __restrict__ h, long n128) {
  long i = (long)blockIdx.x * blockDim.x + threadIdx.x;
  const long stride = (long)gridDim.x * blockDim.x;
  for (; i < n128; i += stride) {
    const long node = i >> 7;
    const int c = (int)(i & 127);
    const float* gin = gi + node * G3;
    const float* ghn = gh + node * G3;
    const float r = 1.0f / (1.0f + __expf(-(gin[c] + ghn[c])));
    const float z = 1.0f / (1.0f + __expf(-(gin[c + HID] + ghn[c + HID])));
    const float nn = tanhf(gin[c + 2 * HID] + r * ghn[c + 2 * HID]);
    h[i] = (1.0f - z) * nn + z * h[i];
  }
}

__device__ __forceinline__ void atomicMaxFloat(float* addr, float v) {
  if (v >= 0.0f) atomicMax((int*)addr, __float_as_int(v));
  else           atomicMin((unsigned int*)addr, (unsigned int)__float_as_int(v));
}

__global__ void pool_max_kernel(const float* __restrict__ h,
                                const int* __restrict__ batch,
                                float* __restrict__ pooled, long n128) {
  long i = (long)blockIdx.x * blockDim.x + threadIdx.x;
  const long stride = (long)gridDim.x * blockDim.x;
  for (; i < n128; i += stride) {
    const long node = i >> 7;
    const int c = (int)(i & 127);
    atomicMaxFloat(&pooled[(size_t)batch[node] * HID + c], h[i]);
  }
}

// Per-graph: finite-guard pooled (write guarded pooled out), then
// z1 = relu(pooled @ W1.T + b1) [64], logit = z1 @ W2.T + b2.
__global__ void classifier_kernel(const float* __restrict__ W1,
                                  const float* __restrict__ b1,
                                  const float* __restrict__ W2,
                                  const float* __restrict__ b2,
                                  float* __restrict__ out, int G) {
  __shared__ float prow[HID];
  __shared__ float z1[HID / 2];
  const int g = blockIdx.x;
  const int t = threadIdx.x;  // 128 threads
  float* pooled = out + G;    // pooled region of d_out (already max-reduced)
  float v = pooled[(size_t)g * HID + t];
  if (!(v > -1.0e38f && v < 1.0e38f)) v = 0.0f;  // -inf (empty graph) / NaN -> 0
  prow[t] = v;
  pooled[(size_t)g * HID + t] = v;
  __syncthreads();
  if (t < HID / 2) {
    float s = b1[t];
    const float* wr = W1 + (size_t)t * HID;
#pragma unroll 8
    for (int k = 0; k < HID; ++k) s += prow[k] * wr[k];
    z1[t] = fmaxf(s, 0.0f);
  }
  __syncthreads();
  if (t == 0) {
    float s = b2[0];
    for (int j = 0; j < HID / 2; ++j) s += z1[j] * W2[j];
    out[g] = s;
  }
}

// ---------------------------------------------------------------------------
extern "C" void kernel_launch(void* const* d_in, const int* in_sizes, int n_in,
                              void* d_out, int out_size, void* d_ws, size_t ws_size,
                              hipStream_t stream) {
  const int*   x_lex   = (const int*)d_in[0];
  const int*   edge    = (const int*)d_in[1];
  const int*   batch   = (const int*)d_in[2];
  const float* table   = (const float*)d_in[3];
  const float* W_in    = (const float*)d_in[4];
  const float* b_in    = (const float*)d_in[5];
  const float* ggnn_W  = (const float*)d_in[6];
  const float* W_ih    = (const float*)d_in[7];
  const float* W_hh    = (const float*)d_in[8];
  const float* b_ih    = (const float*)d_in[9];
  const float* b_hh    = (const float*)d_in[10];
  const float* cls_W1  = (const float*)d_in[11];
  const float* cls_b1  = (const float*)d_in[12];
  const float* cls_W2  = (const float*)d_in[13];
  const float* cls_b2  = (const float*)d_in[14];

  const int N = in_sizes[0];
  const int E = in_sizes[1] / 2;
  const int G = out_size / (1 + HID);  // logits[G] + pooled[G*HID]
  const int NUM_LAYERS = in_sizes[6] / (HID * HID);

  const int* src = edge;
  const int* dst = edge + E;

  // Workspace layout (floats): h | m | agg | gi | gh
  float* h   = (float*)d_ws;
  float* m   = h   + (size_t)N * HID;
  float* agg = m   + (size_t)N * HID;
  float* gi  = agg + (size_t)N * HID;
  float* gh  = gi  + (size_t)N * G3;
  (void)ws_size; (void)n_in;

  float* outp = (float*)d_out;

  const long n128 = (long)N * HID;
  const int rowBlocks = (N + 127) / 128;
  const dim3 blk(256);
  const dim3 grid8(rowBlocks, HID / 64);   // J=128: 2 col blocks
  const dim3 grid24(rowBlocks, G3 / 64);   // J=384: 6 col blocks
  const int ewGrid = (int)((n128 + 255) / 256);
  const int scatGrid = (E + 7) / 8;        // one wave per edge, 8 waves/block

  // 1) h = embed_table[x_lex] @ W_in.T + b_in   (gather fused into A-frags)
  wmma_gemm_kernel<true, true><<<grid8, blk, 0, stream>>>(
      table, x_lex, W_in, b_in, h, N, HID);

  // 2) GGNN layers
  for (int l = 0; l < NUM_LAYERS; ++l) {
    wmma_gemm_kernel<false, false><<<grid8, blk, 0, stream>>>(
        h, nullptr, ggnn_W + (size_t)l * HID * HID, nullptr, m, N, HID);
    fill_kernel<<<ewGrid, blk, 0, stream>>>(agg, 0.0f, n128);
    scatter_add_kernel<<<scatGrid, blk, 0, stream>>>(m, src, dst, agg, E);
    wmma_gemm_kernel<true, false><<<grid24, blk, 0, stream>>>(
        agg, nullptr, W_ih, b_ih, gi, N, G3);
    wmma_gemm_kernel<true, false><<<grid24, blk, 0, stream>>>(
        h, nullptr, W_hh, b_hh, gh, N, G3);
    gru_gate_kernel<<<ewGrid, blk, 0, stream>>>(gi, gh, h, n128);
  }

  // 3) global max pool per graph into d_out pooled region
  fill_kernel<<<(G * HID + 255) / 256, blk, 0, stream>>>(
      outp + G, -__builtin_inff(), (long)G * HID);
  pool_max_kernel<<<ewGrid, blk, 0, stream>>>(h, batch, outp + G, n128);

  // 4) finite-guard + classifier head
  classifier_kernel<<<G, dim3(HID), 0, stream>>>(
      cls_W1, cls_b1, cls_W2, cls_b2, outp, G);
}